// Laplacian1DLinearAttention_60559038873885
// MI455X (gfx1250) — compile-verified
//
#include <hip/hip_runtime.h>
#include <hip/hip_bf16.h>

#define B_    2
#define N_    1024
#define C_    512
#define H_    8
#define D_    64
#define M_    128          // pooled length N/POOL
#define POOL_ 8
#define QKV_  1536

typedef float v2f __attribute__((ext_vector_type(2)));
typedef float v8f __attribute__((ext_vector_type(8)));

// ---------------------------------------------------------------------------
// fp32 WMMA register-tiled K-loop: MT x NT tiles of 16x16, A fragments reused
// across NT column tiles, B fragments reused across MT row tiles.
// Layouts per CDNA5 ISA 7.12.2:
//   A (16x4):  lane = 16*half + m ; a[v] = A[m][k0 + 2*half + v]
//   B (4x16):  lane = 16*half + n ; b[v] = B[k0 + 2*half + v][n]
//   C/D:       row  = v + 8*half  ; col = lane & 15
// ---------------------------------------------------------------------------
template <int MT, int NT, typename FA, typename FB>
__device__ __forceinline__ void wmma_f32_tiles(int K, FA loadA, FB loadB,
                                               v8f (&acc)[MT][NT]) {
  const int lane = threadIdx.x & 31;
  const int half = lane >> 4;
  const int sub  = lane & 15;
  for (int k0 = 0; k0 < K; k0 += 4) {
    const int ka = k0 + 2 * half;
    v2f a[MT], b[NT];
#pragma unroll
    for (int i = 0; i < MT; ++i) {
      a[i].x = loadA(i * 16 + sub, ka);
      a[i].y = loadA(i * 16 + sub, ka + 1);
    }
#pragma unroll
    for (int j = 0; j < NT; ++j) {
      b[j].x = loadB(ka, j * 16 + sub);
      b[j].y = loadB(ka + 1, j * 16 + sub);
    }
#pragma unroll
    for (int i = 0; i < MT; ++i)
#pragma unroll
      for (int j = 0; j < NT; ++j)
        acc[i][j] = __builtin_amdgcn_wmma_f32_16x16x4_f32(
            false, a[i], false, b[j], (short)0, acc[i][j], false, false);
  }
}

// ---------------------------------------------------------------------------
// 1) qkv = (x @ qkv_w^T + qkv_b) * mask      out: (B*N, 1536)
//    32x64 tile per wave (2x4 accumulators = 64 VGPRs, no spills).
// ---------------------------------------------------------------------------
__global__ void __launch_bounds__(256, 1)
k_gemm_qkv(const float* __restrict__ x, const float* __restrict__ w,
           const float* __restrict__ bias, const float* __restrict__ tm,
           float* __restrict__ out) {
  const int wave = (blockIdx.x * blockDim.x + threadIdx.x) >> 5;
  const int lane = threadIdx.x & 31;
  const int half = lane >> 4, sub = lane & 15;
  const int mt = wave / (QKV_ / 64);   // 64 row tiles of 32
  const int nt = wave % (QKV_ / 64);   // 24 col tiles of 64
  const int m0 = mt * 32, n0 = nt * 64;
  __builtin_prefetch(w + (size_t)n0 * C_, 0, 1);
  __builtin_prefetch(x + (size_t)m0 * C_, 0, 1);
  v8f acc[2][4] = {};
  wmma_f32_tiles<2, 4>(
      C_, [&](int r, int k) { return x[(size_t)(m0 + r) * C_ + k]; },
      [&](int k, int c) { return w[(size_t)(n0 + c) * C_ + k]; }, acc);
#pragma unroll
  for (int j = 0; j < 4; ++j) {
    const int col = n0 + j * 16 + sub;
    const float bv = bias[col];
#pragma unroll
    for (int i = 0; i < 2; ++i)
#pragma unroll
      for (int v = 0; v < 8; ++v) {
        const int row = m0 + i * 16 + v + 8 * half;
        out[(size_t)row * QKV_ + col] = (acc[i][j][v] + bv) * tm[row];
      }
  }
}

// ---------------------------------------------------------------------------
// 2) v_local via depthwise conv on the SCRAMBLED (B,C,N) view of v.
//    v_sp[b][r][s] = qkv[b][flat>>9][1024 + (flat&511)],  flat = r*1024+s
// ---------------------------------------------------------------------------
__global__ void k_dwconv(const float* __restrict__ qkv, const float* __restrict__ dwc_w,
                         const float* __restrict__ dwc_b, const float* __restrict__ tm,
                         float* __restrict__ vlocal) {
  const int i = blockIdx.x * blockDim.x + threadIdx.x;  // [0, B*C*N)
  if (i >= B_ * C_ * N_) return;
  const int s = i % N_;
  const int r = (i / N_) % C_;
  const int b = i / (N_ * C_);
  float acc = 0.f;
#pragma unroll
  for (int t = 0; t < 3; ++t) {
    const int sp = s + t - 1;
    if (sp >= 0 && sp < N_) {
      const int flat = r * N_ + sp;
      const int n = flat >> 9;
      const int cc = flat & (C_ - 1);
      acc += dwc_w[r * 3 + t] * qkv[(size_t)(b * N_ + n) * QKV_ + 2 * C_ + cc];
    }
  }
  vlocal[(size_t)(b * N_ + s) * C_ + r] = (acc + dwc_b[r]) * tm[b * N_ + s];
}

// ---------------------------------------------------------------------------
// 3) masked 8x pooling on the scrambled view -> q_l/k_l (B,H,m,D), lmask
// ---------------------------------------------------------------------------
__global__ void k_pool(const float* __restrict__ qkv, const float* __restrict__ tm,
                       float* __restrict__ ql, float* __restrict__ kl,
                       float* __restrict__ lmask) {
  const int i = blockIdx.x * blockDim.x + threadIdx.x;  // [0, B*C*M)
  if (i >= B_ * C_ * M_) return;
  const int mi = i % M_;
  const int c = (i / M_) % C_;
  const int b = i / (M_ * C_);
  float pmsum = 0.f, qa = 0.f, ka = 0.f;
#pragma unroll
  for (int j = 0; j < POOL_; ++j) {
    const int nn = mi * POOL_ + j;
    const float mv = tm[b * N_ + nn];
    pmsum += mv;
    const int flat = c * N_ + nn;
    const int n = flat >> 9;
    const int cc = flat & (C_ - 1);
    const float* row = qkv + (size_t)(b * N_ + n) * QKV_;
    qa += row[cc] * mv;
    ka += row[C_ + cc] * mv;
  }
  const float pm = pmsum * (1.f / POOL_);
  const float inv = 1.f / (POOL_ * fmaxf(pm, 1e-6f));
  const int h = c >> 6, d = c & 63;
  const int bh = b * H_ + h;
  ql[((size_t)bh * M_ + mi) * D_ + d] = qa * inv;
  kl[((size_t)bh * M_ + mi) * D_ + d] = ka * inv;
  if (c == 0) lmask[b * M_ + mi] = (pm > 0.f) ? 1.f : 0.f;
}

// ---------------------------------------------------------------------------
// 4) W_eps[b,h,i,j] = exp(-L1/4)*pair + I*(1-pair) + 1e-4*I
// ---------------------------------------------------------------------------
__global__ void k_weps(const float* __restrict__ ql, const float* __restrict__ kl,
                       const float* __restrict__ lmask, float* __restrict__ weps) {
  const int i = blockIdx.x * blockDim.x + threadIdx.x;  // [0, 16*128*128)
  if (i >= B_ * H_ * M_ * M_) return;
  const int j = i & (M_ - 1);
  const int ii = (i >> 7) & (M_ - 1);
  const int bh = i >> 14;
  const int b = bh >> 3;
  const float* qr = ql + ((size_t)bh * M_ + ii) * D_;
  const float* kr = kl + ((size_t)bh * M_ + j) * D_;
  float l1 = 0.f;
  for (int d = 0; d < D_; ++d) l1 += fabsf(qr[d] - kr[d]);
  const float pair = lmask[b * M_ + ii] * lmask[b * M_ + j];
  float v = expf(-l1 * 0.25f) * pair;
  if (ii == j) v += (1.f - pair) + 1e-4f;
  weps[i] = v;
}

// ---------------------------------------------------------------------------
// 5) X0 = 2/(||W_eps||_F + 1e-8) * W_eps^T      one block per (b,h)
// ---------------------------------------------------------------------------
__global__ void k_ns_init(const float* __restrict__ weps, float* __restrict__ x0) {
  __shared__ float red[256];
  const int bh = blockIdx.x;
  const float* W = weps + (size_t)bh * M_ * M_;
  float s = 0.f;
  for (int i = threadIdx.x; i < M_ * M_; i += 256) {
    const float w = W[i];
    s += w * w;
  }
  red[threadIdx.x] = s;
  __syncthreads();
  for (int off = 128; off > 0; off >>= 1) {
    if (threadIdx.x < off) red[threadIdx.x] += red[threadIdx.x + off];
    __syncthreads();
  }
  const float scale = 2.f / (sqrtf(red[0]) + 1e-8f);
  float* X = x0 + (size_t)bh * M_ * M_;
  for (int o = threadIdx.x; o < M_ * M_; o += 256) {
    const int i = o >> 7, j = o & (M_ - 1);
    X[o] = scale * W[j * M_ + i];
  }
}

// ---------------------------------------------------------------------------
// 6a) T = 2I - W_eps @ X  (16x16 tiles: 1024 waves, NS chain is latency-bound)
// ---------------------------------------------------------------------------
__global__ void __launch_bounds__(256, 1)
k_ns_t(const float* __restrict__ weps, const float* __restrict__ X,
       float* __restrict__ T) {
  const int wave = (blockIdx.x * blockDim.x + threadIdx.x) >> 5;
  const int lane = threadIdx.x & 31;
  const int half = lane >> 4, sub = lane & 15;
  const int bh = wave >> 6;
  const int it = (wave >> 3) & 7;
  const int jt = wave & 7;
  const float* Wm = weps + (size_t)bh * M_ * M_;
  const float* Xm = X + (size_t)bh * M_ * M_;
  v8f acc[1][1] = {};
  wmma_f32_tiles<1, 1>(
      M_, [&](int r, int k) { return Wm[(it * 16 + r) * M_ + k]; },
      [&](int k, int c) { return Xm[k * M_ + jt * 16 + c]; }, acc);
  float* Tm = T + (size_t)bh * M_ * M_;
  const int col = jt * 16 + sub;
#pragma unroll
  for (int v = 0; v < 8; ++v) {
    const int row = it * 16 + v + 8 * half;
    Tm[row * M_ + col] = ((row == col) ? 2.f : 0.f) - acc[0][0][v];
  }
}

// 6b) Xo = X @ T
__global__ void __launch_bounds__(256, 1)
k_ns_x(const float* __restrict__ X, const float* __restrict__ T,
       float* __restrict__ Xo) {
  const int wave = (blockIdx.x * blockDim.x + threadIdx.x) >> 5;
  const int lane = threadIdx.x & 31;
  const int half = lane >> 4, sub = lane & 15;
  const int bh = wave >> 6;
  const int it = (wave >> 3) & 7;
  const int jt = wave & 7;
  const float* Xm = X + (size_t)bh * M_ * M_;
  const float* Tm = T + (size_t)bh * M_ * M_;
  v8f acc[1][1] = {};
  wmma_f32_tiles<1, 1>(
      M_, [&](int r, int k) { return Xm[(it * 16 + r) * M_ + k]; },
      [&](int k, int c) { return Tm[k * M_ + jt * 16 + c]; }, acc);
  float* Om = Xo + (size_t)bh * M_ * M_;
  const int col = jt * 16 + sub;
#pragma unroll
  for (int v = 0; v < 8; ++v) Om[(it * 16 + v + 8 * half) * M_ + col] = acc[0][0][v];
}

// ---------------------------------------------------------------------------
// 7) C_q / C_k :  out[b,h,n,j] = exp(-L1(q[n], cen[j])/4) * lmask[b,j]
//    centroids staged in LDS (32 KB), 128 threads = j, 8 rows n per block
// ---------------------------------------------------------------------------
__global__ void k_lap_cross(const float* __restrict__ qkv, int col_off,
                            const float* __restrict__ cen,
                            const float* __restrict__ lmask,
                            float* __restrict__ out) {
  __shared__ float ldsC[M_ * D_];
  __shared__ float qrow[D_];
  const int bh = blockIdx.x >> 7;   // grid.x = 16*128
  const int nch = blockIdx.x & 127;
  const int b = bh >> 3, h = bh & 7;
  const int j = threadIdx.x;        // 128 threads
  for (int d = 0; d < D_; ++d) ldsC[j * D_ + d] = cen[((size_t)bh * M_ + j) * D_ + d];
  __syncthreads();
  const float lm = lmask[b * M_ + j];
  for (int ns = 0; ns < 8; ++ns) {
    const int n = nch * 8 + ns;
    const float* q = qkv + (size_t)(b * N_ + n) * QKV_ + col_off + h * D_;
    if (threadIdx.x < D_) qrow[threadIdx.x] = q[threadIdx.x];
    __syncthreads();
    float l1 = 0.f;
    for (int d = 0; d < D_; ++d) l1 += fabsf(qrow[d] - ldsC[j * D_ + d]);
    out[((size_t)bh * N_ + n) * M_ + j] = expf(-l1 * 0.25f) * lm;
    __syncthreads();
  }
}

// ---------------------------------------------------------------------------
// 8) whitening stats over (b,n) per (h,j)
// ---------------------------------------------------------------------------
__global__ void k_whiten_stats(const float* __restrict__ cq, const float* __restrict__ tm,
                               float* __restrict__ mu, float* __restrict__ rstd) {
  __shared__ float r0[256], r1[256], r2[256];
  const int hj = blockIdx.x;  // h*128 + j
  const int h = hj >> 7, j = hj & 127;
  float s0 = 0.f, s1 = 0.f, s2 = 0.f;
  for (int i = threadIdx.x; i < B_ * N_; i += 256) {
    const int b = i >> 10, n = i & (N_ - 1);
    const float mv = tm[i];
    const float c = cq[(((size_t)(b * H_ + h)) * N_ + n) * M_ + j];
    s0 += mv;
    s1 += mv * c;
    s2 += mv * c * c;
  }
  r0[threadIdx.x] = s0; r1[threadIdx.x] = s1; r2[threadIdx.x] = s2;
  __syncthreads();
  for (int off = 128; off > 0; off >>= 1) {
    if (threadIdx.x < off) {
      r0[threadIdx.x] += r0[threadIdx.x + off];
      r1[threadIdx.x] += r1[threadIdx.x + off];
      r2[threadIdx.x] += r2[threadIdx.x + off];
    }
    __syncthreads();
  }
  if (threadIdx.x == 0) {
    const float valid = fmaxf(r0[0], 1.f);
    const float m = r1[0] / valid;
    const float var = (r2[0] - 2.f * m * r1[0] + m * m * r0[0]) / valid;
    mu[hj] = m;
    rstd[hj] = rsqrtf(var + 1e-5f);
  }
}

// 9) in-place normalize C_q
__global__ void k_whiten_apply(float* __restrict__ cq, const float* __restrict__ tm,
                               const float* __restrict__ mu,
                               const float* __restrict__ rstd) {
  const size_t i = (size_t)blockIdx.x * blockDim.x + threadIdx.x;
  if (i >= (size_t)B_ * H_ * N_ * M_) return;
  const int j = (int)(i & (M_ - 1));
  const size_t rest = i >> 7;
  const int n = (int)(rest & (N_ - 1));
  const int bh = (int)(rest >> 10);
  const int h = bh & 7, b = bh >> 3;
  const int hj = h * M_ + j;
  cq[i] = (cq[i] - mu[hj]) * rstd[hj] * tm[b * N_ + n];
}

// ---------------------------------------------------------------------------
// 10) ctx1 = C_k^T @ v   (per bh: 128x64, K=1024) 16x64 tile per wave
// ---------------------------------------------------------------------------
__global__ void __launch_bounds__(256, 1)
k_gemm_ckv(const float* __restrict__ ck, const float* __restrict__ qkv,
           float* __restrict__ ctx1) {
  const int wave = (blockIdx.x * blockDim.x + threadIdx.x) >> 5;
  const int lane = threadIdx.x & 31;
  const int half = lane >> 4, sub = lane & 15;
  const int bh = wave >> 3;
  const int jt = wave & 7;
  const int b = bh >> 3, h = bh & 7;
  const float* ckm = ck + (size_t)bh * N_ * M_;
  const float* vbase = qkv + 2 * C_ + h * D_;
  v8f acc[1][4] = {};
  wmma_f32_tiles<1, 4>(
      N_, [&](int r, int k) { return ckm[(size_t)k * M_ + jt * 16 + r]; },
      [&](int k, int c) { return vbase[(size_t)(b * N_ + k) * QKV_ + c]; }, acc);
  float* out = ctx1 + (size_t)bh * M_ * D_;
#pragma unroll
  for (int j = 0; j < 4; ++j)
#pragma unroll
    for (int v = 0; v < 8; ++v)
      out[(jt * 16 + v + 8 * half) * D_ + j * 16 + sub] = acc[0][j][v];
}

// 11) ctx = W_inv @ ctx1   (per bh: 128x64, K=128) 16x64 tile per wave
__global__ void __launch_bounds__(256, 1)
k_gemm_winv(const float* __restrict__ winv, const float* __restrict__ ctx1,
            float* __restrict__ ctx) {
  const int wave = (blockIdx.x * blockDim.x + threadIdx.x) >> 5;
  const int lane = threadIdx.x & 31;
  const int half = lane >> 4, sub = lane & 15;
  const int bh = wave >> 3;
  const int it = wave & 7;
  const float* Wm = winv + (size_t)bh * M_ * M_;
  const float* Cm = ctx1 + (size_t)bh * M_ * D_;
  v8f acc[1][4] = {};
  wmma_f32_tiles<1, 4>(
      M_, [&](int r, int k) { return Wm[(it * 16 + r) * M_ + k]; },
      [&](int k, int c) { return Cm[k * D_ + c]; }, acc);
  float* out = ctx + (size_t)bh * M_ * D_;
#pragma unroll
  for (int j = 0; j < 4; ++j)
#pragma unroll
    for (int v = 0; v < 8; ++v)
      out[(it * 16 + v + 8 * half) * D_ + j * 16 + sub] = acc[0][j][v];
}

// 12) pre[b,n,h*64+d] = (C_qn @ ctx)[n,d]*tm[b,n] + v_local   16x64 per wave
__global__ void __launch_bounds__(256, 1)
k_gemm_ga(const float* __restrict__ cqn, const float* __restrict__ ctx,
          const float* __restrict__ tm, float* __restrict__ pre) {
  const int wave = (blockIdx.x * blockDim.x + threadIdx.x) >> 5;
  const int lane = threadIdx.x & 31;
  const int half = lane >> 4, sub = lane & 15;
  const int bh = wave >> 6;
  const int nt = wave & 63;
  const int b = bh >> 3, h = bh & 7;
  const float* Am = cqn + (size_t)bh * N_ * M_;
  const float* Bm = ctx + (size_t)bh * M_ * D_;
  v8f acc[1][4] = {};
  wmma_f32_tiles<1, 4>(
      M_, [&](int r, int k) { return Am[(size_t)(nt * 16 + r) * M_ + k]; },
      [&](int k, int c) { return Bm[k * D_ + c]; }, acc);
#pragma unroll
  for (int j = 0; j < 4; ++j) {
    const int col = h * D_ + j * 16 + sub;
#pragma unroll
    for (int v = 0; v < 8; ++v) {
      const int n = nt * 16 + v + 8 * half;
      const size_t addr = (size_t)(b * N_ + n) * C_ + col;
      pre[addr] = acc[0][j][v] * tm[b * N_ + n] + pre[addr];
    }
  }
}

// 13) out = (pre @ proj_w^T + proj_b) * tm     32x64 tile per wave
__global__ void __launch_bounds__(256, 1)
k_gemm_proj(const float* __restrict__ pre, const float* __restrict__ w,
            const float* __restrict__ bias, const float* __restrict__ tm,
            float* __restrict__ out) {
  const int wave = (blockIdx.x * blockDim.x + threadIdx.x) >> 5;
  const int lane = threadIdx.x & 31;
  const int half = lane >> 4, sub = lane & 15;
  const int mt = wave / (C_ / 64);   // 64 row tiles of 32
  const int nt = wave % (C_ / 64);   // 8 col tiles of 64
  const int m0 = mt * 32, n0 = nt * 64;
  __builtin_prefetch(w + (size_t)n0 * C_, 0, 1);
  v8f acc[2][4] = {};
  wmma_f32_tiles<2, 4>(
      C_, [&](int r, int k) { return pre[(size_t)(m0 + r) * C_ + k]; },
      [&](int k, int c) { return w[(size_t)(n0 + c) * C_ + k]; }, acc);
#pragma unroll
  for (int j = 0; j < 4; ++j) {
    const int col = n0 + j * 16 + sub;
    const float bv = bias[col];
#pragma unroll
    for (int i = 0; i < 2; ++i)
#pragma unroll
      for (int v = 0; v < 8; ++v) {
        const int row = m0 + i * 16 + v + 8 * half;
        out[(size_t)row * C_ + col] = (acc[i][j][v] + bv) * tm[row];
      }
  }
}

// ---------------------------------------------------------------------------
extern "C" void kernel_launch(void* const* d_in, const int* in_sizes, int n_in,
                              void* d_out, int out_size, void* d_ws, size_t ws_size,
                              hipStream_t stream) {
  const float* x      = (const float*)d_in[0];
  const float* tm     = (const float*)d_in[1];
  const float* qkv_w  = (const float*)d_in[2];
  const float* qkv_b  = (const float*)d_in[3];
  const float* proj_w = (const float*)d_in[4];
  const float* proj_b = (const float*)d_in[5];
  const float* dwc_w  = (const float*)d_in[6];
  const float* dwc_b  = (const float*)d_in[7];
  float* out = (float*)d_out;

  float* p = (float*)d_ws;
  float* qkv   = p; p += (size_t)B_ * N_ * QKV_;
  float* pre   = p; p += (size_t)B_ * N_ * C_;   // v_local, then ga+v_local
  float* ql    = p; p += (size_t)B_ * H_ * M_ * D_;
  float* kl    = p; p += (size_t)B_ * H_ * M_ * D_;
  float* lmask = p; p += B_ * M_;
  float* weps  = p; p += (size_t)B_ * H_ * M_ * M_;
  float* Xa    = p; p += (size_t)B_ * H_ * M_ * M_;
  float* Xb    = p; p += (size_t)B_ * H_ * M_ * M_;
  float* T     = p; p += (size_t)B_ * H_ * M_ * M_;
  float* cq    = p; p += (size_t)B_ * H_ * N_ * M_;
  float* ck    = p; p += (size_t)B_ * H_ * N_ * M_;
  float* mu    = p; p += H_ * M_;
  float* rstd  = p; p += H_ * M_;
  float* ctx1  = p; p += (size_t)B_ * H_ * M_ * D_;
  float* ctx   = p; p += (size_t)B_ * H_ * M_ * D_;

  // 64x24 = 1536 waves of 32x64 tiles -> 192 blocks of 8 waves
  k_gemm_qkv<<<192, 256, 0, stream>>>(x, qkv_w, qkv_b, tm, qkv);
  k_dwconv<<<4096, 256, 0, stream>>>(qkv, dwc_w, dwc_b, tm, pre);
  k_pool<<<512, 256, 0, stream>>>(qkv, tm, ql, kl, lmask);
  k_weps<<<1024, 256, 0, stream>>>(ql, kl, lmask, weps);
  k_ns_init<<<16, 256, 0, stream>>>(weps, Xa);
  float* Xcur = Xa;
  float* Xnext = Xb;
  for (int it = 0; it < 5; ++it) {
    k_ns_t<<<128, 256, 0, stream>>>(weps, Xcur, T);
    k_ns_x<<<128, 256, 0, stream>>>(Xcur, T, Xnext);
    float* tswap = Xcur; Xcur = Xnext; Xnext = tswap;
  }
  k_lap_cross<<<2048, 128, 0, stream>>>(qkv, 0, kl, lmask, cq);
  k_lap_cross<<<2048, 128, 0, stream>>>(qkv, C_, ql, lmask, ck);
  k_whiten_stats<<<1024, 256, 0, stream>>>(cq, tm, mu, rstd);
  k_whiten_apply<<<8192, 256, 0, stream>>>(cq, tm, mu, rstd);
  // 16 bh * 8 tiles = 128 waves -> 16 blocks
  k_gemm_ckv<<<16, 256, 0, stream>>>(ck, qkv, ctx1);
  k_gemm_winv<<<16, 256, 0, stream>>>(Xcur, ctx1, ctx);
  // 16 bh * 64 tiles = 1024 waves -> 128 blocks
  k_gemm_ga<<<128, 256, 0, stream>>>(cq, ctx, tm, pre);
  // 64x8 = 512 waves of 32x64 tiles -> 64 blocks
  k_gemm_proj<<<64, 256, 0, stream>>>(pre, proj_w, proj_b, tm, out);
}